// Cheby_57458072485949
// MI455X (gfx1250) — compile-verified
//
#include <hip/hip_runtime.h>
#include <hip/hip_bf16.h>
#include <math.h>

// ---------------------------------------------------------------------------
// ChebNet GCN forward for MI455X (gfx1250, wave32).
// GEMMs use full-precision V_WMMA_F32_16X16X4_F32 (matches f32 reference).
// SpMM uses float4 gathers + global_atomic_add_f32 scatter (all data L2-resident).
// ---------------------------------------------------------------------------

#define N_NODES 50000
#define N_EDGES 800000
#define NFEAT   100
#define NHID    128
#define NCLASS  40
#define NCLASSP 48   // NCLASS padded to multiple of 16 for WMMA N-tiling

typedef __attribute__((ext_vector_type(2))) float v2f;
typedef __attribute__((ext_vector_type(8))) float v8f;

// ---------------------------------------------------------------------------
// SpMM scatter: acc[row[e]][:] += val[e] * src[col[e]][:]
// One thread per (edge, 4-float chunk). Consecutive threads -> consecutive
// chunks of the same source row (coalesced float4 gathers).
// NCH = chunks per row, STRIDE = floats per row (compile-time: cheap div).
// ---------------------------------------------------------------------------
template <int NCH, int STRIDE>
__global__ void spmm_atomic_k(const int* __restrict__ rowi,
                              const int* __restrict__ coli,
                              const float* __restrict__ val,
                              const float* __restrict__ src,
                              float* __restrict__ acc) {
    int t = blockIdx.x * blockDim.x + threadIdx.x;
    const int total = N_EDGES * NCH;
    if (t >= total) return;
    int e  = t / NCH;           // compile-time NCH -> magic-number division
    int ch = t - e * NCH;
    int r = rowi[e];
    int c = coli[e];
    float v = val[e];
    const float4 xv =
        reinterpret_cast<const float4*>(src + (size_t)c * STRIDE)[ch];
    float* dst = acc + (size_t)r * STRIDE + (ch << 2);
    atomicAdd(dst + 0, v * xv.x);
    atomicAdd(dst + 1, v * xv.y);
    atomicAdd(dst + 2, v * xv.z);
    atomicAdd(dst + 3, v * xv.w);
}

// ---------------------------------------------------------------------------
// WMMA f32 GEMM: C = A(50000 x K) @ B(K x NSTR) + bias, optional ReLU,
// optional dual store (C0 and C1).
// Grid: 3125 blocks (one 16-row M tile each). blockDim = (NSTR/16) waves * 32.
// Each wave owns one 16-col N tile; K-loop in steps of 4 feeding
// V_WMMA_F32_16X16X4_F32.
//
// VGPR layouts per CDNA5 ISA 7.12.2:
//   A 16x4 f32 : lane (m = lane&15), half-wave selects K-pair {0,1} / {2,3}
//   B 4x16 f32 : lane (n = lane&15), half-wave selects K-pair, vgpr = K lsb
//   C 16x16 f32: vgpr i -> row (i + 8*(lane>=16)), col = lane&15
// ---------------------------------------------------------------------------
template <int K, int NSTR, bool RELU, bool DUAL>
__global__ void gemm_wmma_k(const float* __restrict__ A,
                            const float* __restrict__ B,
                            const float* __restrict__ bias,
                            float* __restrict__ C0,
                            float* __restrict__ C1) {
    const int lane  = threadIdx.x & 31;
    const int wave  = threadIdx.x >> 5;        // N tile index
    const int m16   = blockIdx.x << 4;         // M tile base row
    const int l15   = lane & 15;
    const int kHalf = (lane >> 4) << 1;        // 0 (lanes 0-15) or 2 (16-31)
    const int ncol  = (wave << 4) + l15;

    const float* Ap = A + (size_t)(m16 + l15) * K + kHalf; // 8B aligned
    const float* Bp = B + (size_t)kHalf * NSTR + ncol;

    v8f c = {};
    for (int k0 = 0; k0 < K; k0 += 4) {
        const float2 av = *reinterpret_cast<const float2*>(Ap + k0);
        v2f a;
        a[0] = av.x;                 // A[m][k0+kHalf]
        a[1] = av.y;                 // A[m][k0+kHalf+1]
        v2f b;
        b[0] = Bp[(size_t)k0 * NSTR];        // B[k0+kHalf][n]
        b[1] = Bp[(size_t)(k0 + 1) * NSTR];  // B[k0+kHalf+1][n]
        c = __builtin_amdgcn_wmma_f32_16x16x4_f32(
                false, a, false, b, (short)0, c, false, false);
    }

    const int mBase = m16 + ((lane >> 4) << 3);
    const float bv = bias[ncol];
#pragma unroll
    for (int i = 0; i < 8; ++i) {
        float v = c[i] + bv;
        if (RELU) v = fmaxf(v, 0.0f);
        const size_t off = (size_t)(mBase + i) * NSTR + ncol;
        C0[off] = v;
        if (DUAL) C1[off] = v;
    }
}

// ---------------------------------------------------------------------------
// Pad W2 (128x40 -> 128x48, zeros) and b2 (40 -> 48, zeros).
// ---------------------------------------------------------------------------
__global__ void pad_w2_k(const float* __restrict__ W2,
                         const float* __restrict__ b2,
                         float* __restrict__ W2p,
                         float* __restrict__ b2p) {
    int t = blockIdx.x * blockDim.x + threadIdx.x;
    if (t < NHID * NCLASSP) {
        int r = t / NCLASSP;
        int ccol = t - r * NCLASSP;
        W2p[t] = (ccol < NCLASS) ? W2[r * NCLASS + ccol] : 0.0f;
    }
    if (t < NCLASSP) b2p[t] = (t < NCLASS) ? b2[t] : 0.0f;
}

// ---------------------------------------------------------------------------
// Row-wise log_softmax over the 40 valid columns of the padded logits.
// ---------------------------------------------------------------------------
__global__ void logsoftmax_k(const float* __restrict__ z,
                             float* __restrict__ out) {
    int n = blockIdx.x * blockDim.x + threadIdx.x;
    if (n >= N_NODES) return;
    const float* zp = z + (size_t)n * NCLASSP;
    float m = -INFINITY;
#pragma unroll
    for (int j = 0; j < NCLASS; ++j) m = fmaxf(m, zp[j]);
    float s = 0.0f;
#pragma unroll
    for (int j = 0; j < NCLASS; ++j) s += expf(zp[j] - m);
    const float lg = m + logf(s);
    float* op = out + (size_t)n * NCLASS;
#pragma unroll
    for (int j = 0; j < NCLASS; ++j) op[j] = zp[j] - lg;
}

// ---------------------------------------------------------------------------
// Launch sequence (graph-capture safe: stream-ordered only).
// Workspace layout (floats):
//   S1   : 50000*100 = 5,000,000   (x + A@x accumulator)
//   h    : 50000*128 = 6,400,000   (relu hidden)
//   S2   : 50000*128 = 6,400,000   (h + A@h accumulator)
//   outp : 50000*48  = 2,400,000   (padded logits)
//   W2p  : 128*48    = 6,144
//   b2p  : 48
// Total ~80.8 MB.
// ---------------------------------------------------------------------------
extern "C" void kernel_launch(void* const* d_in, const int* in_sizes, int n_in,
                              void* d_out, int out_size, void* d_ws, size_t ws_size,
                              hipStream_t stream) {
    const float* x    = (const float*)d_in[0];
    const int*   arow = (const int*)d_in[1];
    const int*   acol = (const int*)d_in[2];
    const float* aval = (const float*)d_in[3];
    const float* W1   = (const float*)d_in[4];
    const float* b1   = (const float*)d_in[5];
    const float* W2   = (const float*)d_in[6];
    const float* b2   = (const float*)d_in[7];
    float* out = (float*)d_out;

    float* S1   = (float*)d_ws;
    float* h    = S1 + (size_t)N_NODES * NFEAT;
    float* S2   = h + (size_t)N_NODES * NHID;
    float* outp = S2 + (size_t)N_NODES * NHID;
    float* W2p  = outp + (size_t)N_NODES * NCLASSP;
    float* b2p  = W2p + NHID * NCLASSP;

    // S1 = x  (so SpMM atomics accumulate x + A@x in place)
    hipMemcpyAsync(S1, x, (size_t)N_NODES * NFEAT * sizeof(float),
                   hipMemcpyDeviceToDevice, stream);

    // Pad W2/b2 (independent; goes early on the stream)
    pad_w2_k<<<(NHID * NCLASSP + 255) / 256, 256, 0, stream>>>(W2, b2, W2p, b2p);

    // SpMM1: S1 += A @ x   (25 float4 chunks per edge)
    {
        const int total = N_EDGES * (NFEAT / 4);
        spmm_atomic_k<NFEAT / 4, NFEAT>
            <<<(total + 255) / 256, 256, 0, stream>>>(arow, acol, aval, x, S1);
    }

    // GEMM1: h = relu(S1 @ W1 + b1); dual-store into S2 as well
    gemm_wmma_k<NFEAT, NHID, true, true>
        <<<N_NODES / 16, (NHID / 16) * 32, 0, stream>>>(S1, W1, b1, h, S2);

    // SpMM2: S2 += A @ h   (32 float4 chunks per edge)
    {
        const int total = N_EDGES * (NHID / 4);
        spmm_atomic_k<NHID / 4, NHID>
            <<<(total + 255) / 256, 256, 0, stream>>>(arow, acol, aval, h, S2);
    }

    // GEMM2: outp = S2 @ W2p + b2p (padded N=48)
    gemm_wmma_k<NHID, NCLASSP, false, false>
        <<<N_NODES / 16, (NCLASSP / 16) * 32, 0, stream>>>(S2, W2p, b2p, outp,
                                                           nullptr);

    // log_softmax over 40 classes -> d_out (50000 x 40)
    logsoftmax_k<<<(N_NODES + 255) / 256, 256, 0, stream>>>(outp, out);
}